// Attention_24343874633947
// MI455X (gfx1250) — compile-verified
//
#include <hip/hip_runtime.h>
#include <hip/hip_bf16.h>

// out[j] = sum_n ne_nodes[n][j]  (softmax over a [N,1] tensor's axis=1 is all-ones,
// so the whole attention MLP is dead code; only ne_nodes affects the output).
//
// Memory-bound: 204.8 MB streamed once (~8.8 us at 23.3 TB/s). Expressed as
// ones[16x4] x B[4x16] fp32 WMMA accumulation (v_wmma_f32_16x16x4_f32), followed
// by a two-level deterministic cross-block reduction tree (625 = 25 x 25) so no
// single-WGP pass becomes latency-bound.

typedef __attribute__((ext_vector_type(2))) float v2f;
typedef __attribute__((ext_vector_type(8))) float v8f;

#define HIDDEN 256
#define NROWS  200000
#define NB     625            // 625 * 320 == 200000 exactly
#define ROWS_PER_BLOCK 320
#define NB1    25             // 625 = 25 * 25

__global__ __launch_bounds__(256) void colsum_wmma(const float* __restrict__ ne,
                                                   float* __restrict__ partial) {
    const int tid  = threadIdx.x;
    const int wave = tid >> 5;          // 0..7
    const int lane = tid & 31;
    const int half = lane >> 4;         // 0: K rows {r, r+1}, 1: K rows {r+2, r+3}
    const int n    = lane & 15;         // column within 16-wide group
    const int b    = blockIdx.x;

    // A-matrix: 16x4 all ones (2 VGPRs per lane). Sum over K is invariant to the
    // exact K lane/VGPR striping, so this is layout-robust.
    v2f a; a.x = 1.0f; a.y = 1.0f;
    v8f c0 = {}; v8f c1 = {};

    const int cb0 = wave * 32;          // this wave's first 16-col group
    const int cb1 = cb0 + 16;           // second group (adjacent -> 128B lines)

    const float* base = ne + (size_t)b * ROWS_PER_BLOCK * HIDDEN;

    #pragma unroll 2
    for (int r = 0; r < ROWS_PER_BLOCK; r += 4) {
        // B operand (4x16 fp32): lanes 0-15 carry rows {r, r+1}, lanes 16-31 rows {r+2, r+3}.
        const float* p = base + (size_t)(r + 2 * half) * HIDDEN + n;
        v2f b0, b1;
        b0.x = p[cb0];
        b0.y = p[cb0 + HIDDEN];
        b1.x = p[cb1];
        b1.y = p[cb1 + HIDDEN];
        c0 = __builtin_amdgcn_wmma_f32_16x16x4_f32(false, a, false, b0,
                                                   (short)0, c0, false, false);
        c1 = __builtin_amdgcn_wmma_f32_16x16x4_f32(false, a, false, b1,
                                                   (short)0, c1, false, false);
    }

    // All 16 rows of D are identical column sums; row 0 lives in c[0] with N = lane&15.
    // Lanes 0-15 and 16-31 hold duplicates, so let the low half store.
    if (lane < 16) {
        partial[(size_t)b * HIDDEN + cb0 + n] = c0[0];
        partial[(size_t)b * HIDDEN + cb1 + n] = c1[0];
    }
}

// Level 1: 25 blocks, block g sums partial rows [25g, 25g+25) -> mid[g][j].
__global__ __launch_bounds__(256) void reduce_l1(const float* __restrict__ partial,
                                                 float* __restrict__ mid) {
    const int j = threadIdx.x;          // 0..255, one column per thread
    const int g = blockIdx.x;           // 0..24
    const float* p = partial + (size_t)g * NB1 * HIDDEN + j;
    float s = 0.0f;
    #pragma unroll
    for (int bb = 0; bb < NB1; ++bb)    // fixed order -> bitwise deterministic
        s += p[(size_t)bb * HIDDEN];
    mid[(size_t)g * HIDDEN + j] = s;
}

// Level 2: 1 block sums the 25 intermediate rows -> out[j].
__global__ __launch_bounds__(256) void reduce_l2(const float* __restrict__ mid,
                                                 float* __restrict__ out) {
    const int j = threadIdx.x;
    float s = 0.0f;
    #pragma unroll
    for (int g = 0; g < NB1; ++g)       // fixed order -> bitwise deterministic
        s += mid[(size_t)g * HIDDEN + j];
    out[j] = s;
}

extern "C" void kernel_launch(void* const* d_in, const int* in_sizes, int n_in,
                              void* d_out, int out_size, void* d_ws, size_t ws_size,
                              hipStream_t stream) {
    // setup_inputs order: this_node, relations, ne_nodes, W1, b1, W2, b2
    const float* ne = (const float*)d_in[2];
    float* partial  = (float*)d_ws;                         // 625*256 floats = 640 KB
    float* mid      = partial + (size_t)NB * HIDDEN;        // +25*256 floats = 25.6 KB

    colsum_wmma<<<NB, 256, 0, stream>>>(ne, partial);
    reduce_l1<<<NB1, 256, 0, stream>>>(partial, mid);
    reduce_l2<<<1, 256, 0, stream>>>(mid, (float*)d_out);
}